// CenterLoss_76759655514706
// MI455X (gfx1250) — compile-verified
//
#include <hip/hip_runtime.h>

// ---------------------------------------------------------------------------
// CenterLoss on MI455X (gfx1250), wave32.
//
// loss = ( sum_i clip(||f_i||^2 + ||c_{t_i}||^2 - 2 f_i.c_{t_i}, 1e-12, 1e12)
//          + BATCH*(NUM_CLASSES-1)*1e-12 ) / BATCH
//
// Each wave handles a 16-row tile. Cross terms f_i . c_{t_i} come from the
// diagonal of V_WMMA_F32_16X16X4_F32 accumulation over K=512 (full f32
// precision). Bandwidth-bound: ~20 MB of reads, ~1 us at 23.3 TB/s.
// ---------------------------------------------------------------------------

typedef float v2f __attribute__((ext_vector_type(2)));
typedef float v4f __attribute__((ext_vector_type(4)));
typedef float v8f __attribute__((ext_vector_type(8)));

#define NUM_CLASSES 2048
#define FEAT_DIM    512
#define BATCH       8192

#define ROWS_PER_WAVE   16
#define WAVES_TOTAL     (BATCH / ROWS_PER_WAVE)        // 512
#define THREADS_PER_BLK 256                            // 8 waves (wave32)
#define WAVES_PER_BLK   (THREADS_PER_BLK / 32)
#define NUM_BLOCKS      (WAVES_TOTAL / WAVES_PER_BLK)  // 64

__global__ __launch_bounds__(THREADS_PER_BLK)
void center_loss_main(const float* __restrict__ features,
                      const float* __restrict__ centers,
                      const int*   __restrict__ target,
                      float*       __restrict__ partials) {
  const int lane        = threadIdx.x & 31;
  const int waveInBlock = threadIdx.x >> 5;
  const int waveId      = blockIdx.x * WAVES_PER_BLK + waveInBlock;
  const int row0        = waveId * ROWS_PER_WAVE;

  // WMMA 16x16x4 f32 operand layout (wave32):
  //   A (16x4): lanes 0-15 hold row M=lane, K={0,1} in vgpr{0,1};
  //             lanes 16-31 hold row M=lane-16, K={2,3}.
  //   B (4x16): identical pattern with lane = column N.
  // Diagonal sum is K-permutation invariant, so per 8-K chunk each lane
  // loads one float4 and feeds two WMMAs; A and B use the same index math.
  const int m     = lane & 15;          // tile row (A) == tile column (B)
  const int khalf = (lane >> 4) * 4;    // 0 for lanes 0-15, 4 for lanes 16-31
  const int row   = row0 + m;

  const float* __restrict__ frow = features + (size_t)row * FEAT_DIM;
  const float* __restrict__ crow = centers  + (size_t)target[row] * FEAT_DIM;

  v8f   acc = {};                       // 16x16 f32 accumulator (8 VGPRs)
  float fsq = 0.0f;                     // partial ||f||^2 (this lane's K's)
  float csq = 0.0f;                     // partial ||c||^2

  #pragma unroll 4
  for (int k0 = 0; k0 < FEAT_DIM; k0 += 8) {
    v4f a = *(const v4f*)(frow + k0 + khalf);   // global_load_b128
    v4f b = *(const v4f*)(crow + k0 + khalf);   // global_load_b128 (L2-hot)

    v2f a0 = { a.x, a.y };  v2f b0 = { b.x, b.y };
    v2f a1 = { a.z, a.w };  v2f b1 = { b.z, b.w };

    acc = __builtin_amdgcn_wmma_f32_16x16x4_f32(false, a0, false, b0,
                                                (short)0, acc, false, false);
    acc = __builtin_amdgcn_wmma_f32_16x16x4_f32(false, a1, false, b1,
                                                (short)0, acc, false, false);

    fsq += a.x * a.x + a.y * a.y + a.z * a.z + a.w * a.w;
    csq += b.x * b.x + b.y * b.y + b.z * b.z + b.w * b.w;
  }

  // Lane l and lane l^16 cover complementary K halves of the same row.
  fsq += __shfl_xor(fsq, 16, 32);
  csq += __shfl_xor(csq, 16, 32);

  // D layout: vgpr v -> M=v (lanes 0-15, N=lane) / M=v+8 (lanes 16-31).
  // Diagonal j: j<8 -> acc[j] in lane j; j>=8 -> acc[j-8] in lane j+16.
  const int sel = (lane < 8) ? lane
                             : ((lane >= 16 && lane < 24) ? (lane - 16) : -1);
  float dg = 0.0f;
  dg = (sel == 0) ? acc[0] : dg;
  dg = (sel == 1) ? acc[1] : dg;
  dg = (sel == 2) ? acc[2] : dg;
  dg = (sel == 3) ? acc[3] : dg;
  dg = (sel == 4) ? acc[4] : dg;
  dg = (sel == 5) ? acc[5] : dg;
  dg = (sel == 6) ? acc[6] : dg;
  dg = (sel == 7) ? acc[7] : dg;

  // Route row sums to the diagonal-holding lanes:
  // lane j (j<8) keeps row j; lane 16+j needs row 8+j, held by lane 8+j.
  const int src   = (lane < 16) ? lane : (lane - 8);
  const float fsd = __shfl(fsq, src, 32);
  const float csd = __shfl(csq, src, 32);

  float contrib = 0.0f;
  if (sel >= 0) {
    const float d = fsd + csd - 2.0f * dg;
    contrib = fminf(fmaxf(d, 1e-12f), 1e12f);
  }

  // Deterministic wave reduction (inactive lanes contribute 0).
  #pragma unroll
  for (int off = 16; off > 0; off >>= 1)
    contrib += __shfl_xor(contrib, off, 32);

  if (lane == 0) {
    // Masked-out entries: clip(0) = 1e-12 for (NUM_CLASSES-1) entries/row.
    contrib += (float)ROWS_PER_WAVE * (float)(NUM_CLASSES - 1) * 1e-12f;
    partials[waveId] = contrib;
  }
}

__global__ __launch_bounds__(32)
void center_loss_finalize(const float* __restrict__ partials,
                          float*       __restrict__ out) {
  const int lane = threadIdx.x;   // single wave32
  float s = 0.0f;
  #pragma unroll
  for (int i = lane; i < WAVES_TOTAL; i += 32) s += partials[i];
  #pragma unroll
  for (int off = 16; off > 0; off >>= 1) s += __shfl_xor(s, off, 32);
  if (lane == 0) out[0] = s / (float)BATCH;
}

extern "C" void kernel_launch(void* const* d_in, const int* in_sizes, int n_in,
                              void* d_out, int out_size, void* d_ws, size_t ws_size,
                              hipStream_t stream) {
  const float* features = (const float*)d_in[0];   // [8192, 512] f32
  const float* centers  = (const float*)d_in[1];   // [2048, 512] f32
  const int*   target   = (const int*)d_in[2];     // [8192] int
  float*       out      = (float*)d_out;           // scalar loss
  float*       partials = (float*)d_ws;            // 512 floats of scratch

  center_loss_main<<<NUM_BLOCKS, THREADS_PER_BLK, 0, stream>>>(
      features, centers, target, partials);
  center_loss_finalize<<<1, 32, 0, stream>>>(partials, out);

  (void)in_sizes; (void)n_in; (void)out_size; (void)ws_size;
}